// Interpolator_79405355368639
// MI455X (gfx1250) — compile-verified
//
#include <hip/hip_runtime.h>

typedef __attribute__((ext_vector_type(2))) float v2f;
typedef __attribute__((ext_vector_type(8))) float v8f;

#define N_IN      262144
#define N_OUT     393216
#define ROWS      64
#define NTAPS     121
#define CHUNK     48          // outputs per chunk
#define K_STEPS   18          // 72 / 4
#define TILES_PER_ROW 512     // 8192 chunks / 16
#define TOTAL_TILES   (ROWS * TILES_PER_ROW)

// Banded structure of the 48x72 polyphase matrix: active K-step ranges per M-tile.
#define A0_LO 0
#define A0_HI 12   // rows  0..15 touch o in [0, 50]
#define A1_LO 2
#define A1_HI 15   // rows 16..31 touch o in [11, 60]
#define A2_LO 5
#define A2_HI 17   // rows 32..47 touch o in [22, 71]

// C[r][o] for the 48x72 polyphase coefficient matrix:
// out[48c + r] = sum_o C[r][o] * x[32c - 20 + o]
__device__ __forceinline__ float coef(const float* __restrict__ h, int r, int o) {
    int p     = (2 * r) % 3;
    int sigma = 20 + (2 * r) / 3;
    int i     = sigma + 20 - o;     // polyphase tap index
    int j     = p + 3 * i;          // original filter tap
    return (i >= 0 && j <= NTAPS - 1) ? h[j] : 0.0f;
}

__device__ __forceinline__ v8f wmma_f32(v2f a, v2f b, v8f c) {
    // D = A(16x4 f32) * B(4x16 f32) + C(16x16 f32)
    return __builtin_amdgcn_wmma_f32_16x16x4_f32(
        /*neg_a=*/false, a, /*neg_b=*/false, b,
        /*c_mod=*/(short)0, c, /*reuse_a=*/false, /*reuse_b=*/false);
}

// Branch-free bounds-handled load: clamp address (v_cndmask), unconditional
// load, then mask the value. No exec manipulation, no per-element waits.
__device__ __forceinline__ float safe_ld(const float* __restrict__ p, int gi) {
    const bool ok = (unsigned)gi < (unsigned)N_IN;
    const int  ci = ok ? gi : 0;
    const float m = ok ? 1.0f : 0.0f;
    return p[ci] * m;
}

template <bool SAFE>
__device__ __forceinline__ void do_tile(const float* __restrict__ xr_row,
                                        const float* __restrict__ xi_row,
                                        float* __restrict__ out_row,
                                        int c0,
                                        const v2f (&a0)[A0_HI - A0_LO + 1],
                                        const v2f (&a1)[A1_HI - A1_LO + 1],
                                        const v2f (&a2)[A2_HI - A2_LO + 1],
                                        int lane) {
    const int Lc = lane & 15;          // column (chunk) within tile / A row
    const int ko = (lane >> 4) << 1;   // K sub-offset: 0 for lanes 0-15, 2 for 16-31
    // B-fragment base index into this row: element o = 4k + ko of column Lc's window
    const int gbase = 32 * c0 - 20 + 32 * Lc + ko;

    v8f accR0 = {}, accR1 = {}, accR2 = {};
    v8f accI0 = {}, accI1 = {}, accI2 = {};

#pragma unroll
    for (int k = 0; k < K_STEPS; ++k) {
        const int gi = gbase + 4 * k;
        v2f br, bi;
        if (SAFE) {
            br.x = safe_ld(xr_row, gi);
            br.y = safe_ld(xr_row, gi + 1);
            bi.x = safe_ld(xi_row, gi);
            bi.y = safe_ld(xi_row, gi + 1);
        } else {
            br = *(const v2f*)(xr_row + gi);   // 8B-aligned global_load_b64
            bi = *(const v2f*)(xi_row + gi);
        }
        if (k >= A0_LO && k <= A0_HI) {        // compile-time after unroll
            accR0 = wmma_f32(a0[k - A0_LO], br, accR0);
            accI0 = wmma_f32(a0[k - A0_LO], bi, accI0);
        }
        if (k >= A1_LO && k <= A1_HI) {
            accR1 = wmma_f32(a1[k - A1_LO], br, accR1);
            accI1 = wmma_f32(a1[k - A1_LO], bi, accI1);
        }
        if (k >= A2_LO && k <= A2_HI) {
            accR2 = wmma_f32(a2[k - A2_LO], br, accR2);
            accI2 = wmma_f32(a2[k - A2_LO], bi, accI2);
        }
    }

    // D layout: VGPR q -> M = q (lanes 0-15) or q+8 (lanes 16-31); N = lane & 15
    const int col   = c0 + Lc;                 // chunk index
    const int rbase = (lane >> 4) << 3;        // 0 or 8
    const v8f aR[3] = {accR0, accR1, accR2};
    const v8f aI[3] = {accI0, accI1, accI2};
#pragma unroll
    for (int m = 0; m < 3; ++m) {
#pragma unroll
        for (int q = 0; q < 8; q += 2) {
            const int n = CHUNK * col + 16 * m + rbase + q;   // n is even here
            float4 v;
            v.x = aR[m][q];     v.y = aI[m][q];       // out[n]   (re, im)
            v.z = aR[m][q + 1]; v.w = aI[m][q + 1];   // out[n+1] (re, im)
            *(float4*)(out_row + 2 * (size_t)n) = v;  // 16B-aligned store
        }
    }
}

__global__ void __launch_bounds__(256)
resample_wmma_kernel(const float* __restrict__ xr, const float* __restrict__ xi,
                     const float* __restrict__ h, float* __restrict__ out) {
    const int lane  = threadIdx.x & 31;
    const int wave  = blockIdx.x * (blockDim.x >> 5) + (threadIdx.x >> 5);
    const int nwave = gridDim.x * (blockDim.x >> 5);

    // Build only the nonzero A-fragments (16x4 f32 layout) once per wave.
    // A layout: lanes 0-15: M=lane, {K=0,K=1}; lanes 16-31: M=lane-16, {K=2,K=3}.
    v2f a0[A0_HI - A0_LO + 1];
    v2f a1[A1_HI - A1_LO + 1];
    v2f a2[A2_HI - A2_LO + 1];
    const int arow = lane & 15;
    const int ko   = (lane >> 4) << 1;
#pragma unroll
    for (int k = A0_LO; k <= A0_HI; ++k) {
        const int o = 4 * k + ko;
        a0[k - A0_LO].x = coef(h, arow, o);
        a0[k - A0_LO].y = coef(h, arow, o + 1);
    }
#pragma unroll
    for (int k = A1_LO; k <= A1_HI; ++k) {
        const int o = 4 * k + ko;
        a1[k - A1_LO].x = coef(h, 16 + arow, o);
        a1[k - A1_LO].y = coef(h, 16 + arow, o + 1);
    }
#pragma unroll
    for (int k = A2_LO; k <= A2_HI; ++k) {
        const int o = 4 * k + ko;
        a2[k - A2_LO].x = coef(h, 32 + arow, o);
        a2[k - A2_LO].y = coef(h, 32 + arow, o + 1);
    }

    for (int t = wave; t < TOTAL_TILES; t += nwave) {
        const int row = t >> 9;        // / TILES_PER_ROW
        const int ct  = t & (TILES_PER_ROW - 1);
        const float* xr_row = xr + (size_t)row * N_IN;
        const float* xi_row = xi + (size_t)row * N_IN;
        float* out_row      = out + (size_t)row * (2 * (size_t)N_OUT);
        const int c0 = ct << 4;
        if (ct == 0 || ct == TILES_PER_ROW - 1)
            do_tile<true>(xr_row, xi_row, out_row, c0, a0, a1, a2, lane);
        else
            do_tile<false>(xr_row, xi_row, out_row, c0, a0, a1, a2, lane);
    }
}

extern "C" void kernel_launch(void* const* d_in, const int* in_sizes, int n_in,
                              void* d_out, int out_size, void* d_ws, size_t ws_size,
                              hipStream_t stream) {
    const float* x_real = (const float*)d_in[0];   // (64, 262144) f32
    const float* x_imag = (const float*)d_in[1];   // (64, 262144) f32
    const float* h      = (const float*)d_in[2];   // (121,) f32
    float* out = (float*)d_out;                    // (64, 393216) complex64 interleaved

    dim3 grid(1024), block(256);                   // 8192 waves, 4 tiles each
    resample_wmma_kernel<<<grid, block, 0, stream>>>(x_real, x_imag, h, out);
}